// heatsolve_67929202753585
// MI455X (gfx1250) — compile-verified
//
#include <hip/hip_runtime.h>
#include <cstdint>

// ---------------------------------------------------------------------------
// 2D heat equation, 512x512, 400 frames, persistent-kernel temporal blocking.
//
// Roofline: output is 419 MB (mandatory) -> ~18us @ 23.3 TB/s HBM. State per
// step is 1 MB (L2-resident, 192 MB L2). The real costs are (a) cross-step
// sync and (b) LDS bandwidth. We block T=7 steps per device-wide sync
// (399 = 57*7) in LDS (tile 22 rows x 512 cols, ping-pong, 88 KB of the
// 320 KB WGP LDS), so only 56 global barriers instead of 399 (or 400 graph
// launches). Halo reloads use the CDNA5 async global->LDS path (ASYNCcnt).
// Compute pipeline rolls N/C/S rows through registers: 1 b128 + 2 b32 loads
// + 1 b128 store per 4 points on the LDS hot path.
// ---------------------------------------------------------------------------

#define NXG     512
#define RWG     8                      // rows owned per workgroup
#define TCH     7                      // temporal block (sub-steps per chunk)
#define TROWS   (RWG + 2 * TCH)        // 22 tile rows (halo = TCH each side)
#define NWGK    64                     // 512 / RWG workgroups (persistent)
#define NTHR    512                    // 16 waves of 32 (wave32)
#define NGRP    4                      // row groups (NTHR / C4N)
#define ROWS_PG 5                      // contiguous interior rows per group
#define NCHUNKS 57                     // 399 / 7
#define FRAME   (NXG * NXG)
#define C4N     (NXG / 4)              // 128 float4 per row

typedef __attribute__((ext_vector_type(4))) float vf4;   // native vector for NT store

// --- CDNA5 async global->LDS copy (ASYNCcnt path, see CDNA5 ISA) ---
// Single __shared__ array => module LDS base offset 0, so a plain byte offset
// is a valid LDS address operand.
__device__ __forceinline__ void async_load_b128(unsigned lds_byte_off, const void* gsrc) {
    asm volatile("global_load_async_to_lds_b128 %0, %1, off"
                 :: "v"(lds_byte_off),
                    "v"((unsigned long long)(uintptr_t)gsrc)
                 : "memory");
}
#define WAIT_ASYNC0() asm volatile("s_wait_asynccnt 0" ::: "memory")

// --- poison-proof device-wide barrier (epoch equality; 0xAAAAAAAA-safe) -----
__device__ __forceinline__ void grid_barrier(unsigned* flags, int wg, unsigned epoch) {
    WAIT_ASYNC0();
    __builtin_amdgcn_fence(__ATOMIC_RELEASE, "agent");   // drain stores dev-wide
    __syncthreads();
    if (threadIdx.x == 0) {
        __hip_atomic_store(&flags[wg], epoch, __ATOMIC_RELEASE, __HIP_MEMORY_SCOPE_AGENT);
    }
    if (threadIdx.x < NWGK) {
        while (__hip_atomic_load(&flags[threadIdx.x], __ATOMIC_RELAXED,
                                 __HIP_MEMORY_SCOPE_AGENT) != epoch) {
            __builtin_amdgcn_s_sleep(2);
        }
    }
    __builtin_amdgcn_fence(__ATOMIC_ACQUIRE, "agent");   // invalidate stale lines
    __syncthreads();
}

__global__ __launch_bounds__(NTHR, 1)
void heat_persist(const float* __restrict__ u0,
                  const float* __restrict__ alphap,
                  float* __restrict__ out,
                  unsigned* __restrict__ flags)
{
    __shared__ float buf[2][TROWS][NXG];                 // 2 x 44 KB = 88 KB

    const int wg  = (int)blockIdx.x;
    const int tid = (int)threadIdx.x;
    const int r0  = wg * RWG;                            // first owned global row
    const int tb  = r0 - TCH;                            // global row of tile row 0

    const float DXc   = 6.0f / (float)(NXG - 1);
    const float DTc   = 1.0f / 399.0f;
    const float gamma = alphap[0] * (DTc / (DXc * DXc));

    // ---- chunk 0: tile from u0, boundary-clamped (frame-0 semantics) -------
    for (int idx = tid; idx < TROWS * C4N; idx += NTHR) {
        const int i  = idx / C4N;
        const int c4 = idx % C4N;
        const int g  = tb + i;
        float4 v = make_float4(0.f, 0.f, 0.f, 0.f);
        if (g >= 1 && g <= NXG - 2) {
            v = *(const float4*)(u0 + (size_t)g * NXG + (size_t)c4 * 4);
            if (c4 == 0)       v.x = 0.f;
            if (c4 == C4N - 1) v.w = 0.f;
        }
        *(float4*)(&buf[0][i][c4 * 4]) = v;
        // frame 0 = clamped u0: store the owned-row portion straight out
        if (i >= TCH && i < TCH + RWG) {
            *(float4*)(out + (size_t)g * NXG + (size_t)c4 * 4) = v;
        }
    }

    const int c4   = tid & (C4N - 1);
    const int grp  = tid >> 7;                           // 0..3, wave-uniform
    const int ibeg = 1 + grp * ROWS_PG;                  // contiguous rows ibeg..ibeg+4

    int cur = 0;
    for (int chunk = 0; chunk < NCHUNKS; ++chunk) {
        const int t0 = chunk * TCH;

        if (chunk > 0) {
            // Central RWG rows of buf[cur] already hold frame t0 (we computed
            // them). Only the 2*TCH halo rows are refetched -> async
            // global->LDS (ASYNCcnt), rows wave-uniform.
            for (int idx = tid; idx < 2 * TCH * C4N; idx += NTHR) {
                const int h  = idx / C4N;
                const int cc = idx % C4N;
                const int i  = (h < TCH) ? h : (h + RWG);
                const int g  = tb + i;
                if (g >= 0 && g < NXG) {
                    const unsigned lds_off =
                        (unsigned)((((unsigned)cur * TROWS + (unsigned)i) * NXG
                                    + (unsigned)cc * 4u) * sizeof(float));
                    async_load_b128(lds_off,
                                    out + (size_t)t0 * FRAME + (size_t)g * NXG + (size_t)cc * 4);
                }
            }
            WAIT_ASYNC0();
        }
        __syncthreads();

        // ---- T sub-steps, ping-pong LDS, shrinking-validity trapezoid ------
        for (int s = 1; s <= TCH; ++s) {
            const int nxt = cur ^ 1;
            // register-rolled N/C/S pipeline down a contiguous column block
            float4 Nv = *(const float4*)(&buf[cur][ibeg - 1][c4 * 4]);
            float4 Cv = *(const float4*)(&buf[cur][ibeg][c4 * 4]);
            for (int i = ibeg; i < ibeg + ROWS_PG; ++i) {
                const float4 Sv = *(const float4*)(&buf[cur][i + 1][c4 * 4]);
                // w/e reads stay inside the array object even at c4==0/127;
                // their wrong values are masked below.
                const float w = buf[cur][i][c4 * 4 - 1];
                const float e = buf[cur][i][c4 * 4 + 4];
                float4 r;
                r.x = fmaf(gamma, Nv.x + Sv.x + w    + Cv.y - 4.f * Cv.x, Cv.x);
                r.y = fmaf(gamma, Nv.y + Sv.y + Cv.x + Cv.z - 4.f * Cv.y, Cv.y);
                r.z = fmaf(gamma, Nv.z + Sv.z + Cv.y + Cv.w - 4.f * Cv.z, Cv.z);
                r.w = fmaf(gamma, Nv.w + Sv.w + Cv.z + e    - 4.f * Cv.w, Cv.w);
                const int g = tb + i;
                if (g <= 0 || g >= NXG - 1) r = make_float4(0.f, 0.f, 0.f, 0.f);
                if (c4 == 0)       r.x = 0.f;
                if (c4 == C4N - 1) r.w = 0.f;
                *(float4*)(&buf[nxt][i][c4 * 4]) = r;
                // Owned rows stay valid for all s <= TCH (halo = TCH): stream
                // the frame straight from registers. Intermediate frames are
                // never re-read -> non-temporal; chunk-final frame (s==TCH)
                // feeds next chunk's halo reload -> keep it L2-resident.
                if (i >= TCH && i < TCH + RWG) {
                    float* dstp = out + (size_t)(t0 + s) * FRAME
                                      + (size_t)g * NXG + (size_t)c4 * 4;
                    if (s == TCH) {
                        *(float4*)dstp = r;
                    } else {
                        vf4 rv;
                        rv.x = r.x; rv.y = r.y; rv.z = r.z; rv.w = r.w;
                        __builtin_nontemporal_store(rv, (vf4*)dstp);
                    }
                }
                Nv = Cv; Cv = Sv;
            }
            __syncthreads();
            cur = nxt;
        }

        if (chunk + 1 < NCHUNKS) {
            grid_barrier(flags, wg, 0xC0DE0000u + (unsigned)chunk);
        }
    }
    WAIT_ASYNC0();
}

// ---------------------------------------------------------------------------
extern "C" void kernel_launch(void* const* d_in, const int* in_sizes, int n_in,
                              void* d_out, int out_size, void* d_ws, size_t ws_size,
                              hipStream_t stream) {
    (void)in_sizes; (void)n_in; (void)out_size; (void)ws_size;
    const float* u0    = (const float*)d_in[0];
    const float* alpha = (const float*)d_in[1];
    float*       out   = (float*)d_out;
    unsigned*    flags = (unsigned*)d_ws;     // NWGK epoch flags; poison-proof
    heat_persist<<<dim3(NWGK), dim3(NTHR), 0, stream>>>(u0, alpha, out, flags);
}